// RPN_2D_auto_loss_2293512536263
// MI455X (gfx1250) — compile-verified
//
#include <hip/hip_runtime.h>
#include <stdint.h>

// ---- fixed problem config (mirrors reference) ----
#define N_ANCH 276480
#define BATCH  4
#define GSLOTS 32
#define NCLS   4
#define BLK    256
#define NBX    (N_ANCH / BLK)      // 1080, exact
#define NBLK   (NBX * BATCH)       // 4320
#define FG_LO   0.5f
#define IGN_LO  0.5f
#define BEST_LO 0.35f
// kernel B decomposition: 128 blocks x 8 waves x 270 anchors = 276480
#define KB_BLOCKS 128
#define KB_WAVES  8
#define KB_APW    270

typedef unsigned long long u64;
typedef unsigned int       u32;

typedef unsigned int u32x4 __attribute__((ext_vector_type(4)));
typedef int          i32x4 __attribute__((ext_vector_type(4)));
typedef int          i32x8 __attribute__((ext_vector_type(8)));
typedef float        v2f   __attribute__((ext_vector_type(2)));
typedef float        v8f   __attribute__((ext_vector_type(8)));

#ifndef __has_builtin
#define __has_builtin(x) 0
#endif

#if __has_builtin(__builtin_amdgcn_tensor_load_to_lds) && __has_builtin(__builtin_amdgcn_s_wait_tensorcnt)
#define HAVE_TDM 1
#else
#define HAVE_TDM 0
#endif

// ------------------------------------------------------------------
// TDM: DMA a 512B (128 x f32) contiguous tile from global into LDS.
// ------------------------------------------------------------------
#if HAVE_TDM
__device__ __forceinline__ void tdm_load_512B(const void* gsrc, void* lds_dst) {
  u64 ga = (u64)(uintptr_t)gsrc;
  u32 la = (u32)(uintptr_t)lds_dst;      // low 32 bits of flat addr = LDS offset
  u32x4 g0;
  g0.x = 1u;                                            // count=1
  g0.y = la;                                            // lds_addr
  g0.z = (u32)ga;                                       // global_addr lo
  g0.w = (u32)((ga >> 32) & 0x1FFFFFFull) | (2u << 30); // addr hi, type=2
  i32x8 g1;
  g1[0] = (int)(2u << 16);        // data_size=4B
  g1[1] = (int)(128u << 16);      // tensor_dim0 = 128
  g1[2] = (int)(1u << 16);        // tensor_dim1 = 1
  g1[3] = (int)(128u << 16);      // tile_dim0 = 128
  g1[4] = 1;                      // tile_dim1 = 1
  g1[5] = 128;                    // tensor_dim0_stride = 128
  g1[6] = 0;
  g1[7] = 0;
  i32x4 g2; g2[0] = g2[1] = g2[2] = g2[3] = 0;
  i32x4 g3 = g2;
#if defined(__clang_major__) && (__clang_major__ >= 23)
  i32x8 g4; g4[0]=g4[1]=g4[2]=g4[3]=g4[4]=g4[5]=g4[6]=g4[7]=0;
  __builtin_amdgcn_tensor_load_to_lds(g0, g1, g2, g3, g4, 0);
#else
  __builtin_amdgcn_tensor_load_to_lds(g0, g1, g2, g3, 0);
#endif
  __builtin_amdgcn_s_wait_tensorcnt(0);
}
#endif

// xor-swizzle add tree over 32 lanes: ds_swizzle_b32, offset = xor<<10 | 0x1F
__device__ __forceinline__ float swz_add(float v, const int step_xor_imm) {
  (void)step_xor_imm; return v; // unused generic
}
__device__ __forceinline__ float wave_sum32_swz(float v) {
  v += __int_as_float(__builtin_amdgcn_ds_swizzle(__float_as_int(v), 0x401F)); // xor 16
  v += __int_as_float(__builtin_amdgcn_ds_swizzle(__float_as_int(v), 0x201F)); // xor 8
  v += __int_as_float(__builtin_amdgcn_ds_swizzle(__float_as_int(v), 0x101F)); // xor 4
  v += __int_as_float(__builtin_amdgcn_ds_swizzle(__float_as_int(v), 0x081F)); // xor 2
  v += __int_as_float(__builtin_amdgcn_ds_swizzle(__float_as_int(v), 0x041F)); // xor 1
  return v;
}

// Exact f32 sum of 32 lane values, stage 1 via V_WMMA_F32_16X16X4_F32 with
// A = ones: D[m,n] = sum_k B[k,n] = v_n + v_{n+16} (robust to B row striping).
__device__ __forceinline__ float wave_sum_wmma(float v) {
#if __has_builtin(__builtin_amdgcn_wmma_f32_16x16x4_f32)
  v2f a;  a.x = 1.0f; a.y = 1.0f;
  v2f bb; bb.x = v;   bb.y = 0.0f;
  v8f c = {0.f,0.f,0.f,0.f,0.f,0.f,0.f,0.f};
  v8f d = __builtin_amdgcn_wmma_f32_16x16x4_f32(false, a, false, bb,
                                                (short)0, c, false, false);
  float r = d[0];                 // = v[L%16] + v[L%16+16] in every lane
  r += __int_as_float(__builtin_amdgcn_ds_swizzle(__float_as_int(r), 0x201F));
  r += __int_as_float(__builtin_amdgcn_ds_swizzle(__float_as_int(r), 0x101F));
  r += __int_as_float(__builtin_amdgcn_ds_swizzle(__float_as_int(r), 0x081F));
  r += __int_as_float(__builtin_amdgcn_ds_swizzle(__float_as_int(r), 0x041F));
  return r;
#else
  return wave_sum32_swz(v);
#endif
}

// ------------------------------------------------------------------
// Kernel A: zero per-(b,g) best-anchor keys.
// ------------------------------------------------------------------
__global__ void kinit(u64* __restrict__ bestkey) {
  bestkey[threadIdx.x] = 0ull;
}

// ------------------------------------------------------------------
// Kernel B (lane-per-GT): lane g scans a contiguous anchor range with a
// uniform address (scalar s_load path), tracking (best iou, smallest n) with
// zero cross-lane ops; one global_atomic_max_u64 per lane at the end.
// key = (iou_bits<<32) | ~n  ->  ties pick smallest n (jnp.argmax semantics).
// ------------------------------------------------------------------
__global__ __launch_bounds__(BLK) void kbest(
    const float* __restrict__ rois,
    const float* __restrict__ gts_boxes,
    const int*   __restrict__ gts_val,
    u64*         __restrict__ bestkey)
{
  __shared__ float4 sgt[GSLOTS];
  __shared__ int    sval[GSLOTS];

  const int tid  = threadIdx.x;
  const int lane = tid & 31;
  const int b    = blockIdx.y;

  if (tid < GSLOTS) sval[tid] = gts_val[b * GSLOTS + tid];
#if HAVE_TDM
  if (tid < 32) tdm_load_512B(gts_boxes + (size_t)b * GSLOTS * 4, sgt);
#else
  if (tid < GSLOTS * 4)
    ((float*)sgt)[tid] = gts_boxes[(size_t)b * GSLOTS * 4 + tid];
#endif
  __syncthreads();

  // per-lane GT constants
  const float4 gb = sgt[lane];
  const float  ab = (gb.z - gb.x + 1.f) * (gb.w - gb.y + 1.f);
  const bool   vok = sval[lane] > 0;

  const int wvu  = __builtin_amdgcn_readfirstlane(tid >> 5); // uniform wave id
  const int base = (blockIdx.x * KB_WAVES + wvu) * KB_APW;

  float best  = -1.f;
  int   bestn = 0;
#pragma unroll 4
  for (int i = 0; i < KB_APW; ++i) {
    const float4 r = ((const float4*)rois)[base + i];  // uniform address
    float rw = r.z - r.x + 1.f, rh = r.w - r.y + 1.f;
    float ar = rw * rh;
    float x1 = fmaxf(r.x, gb.x), y1 = fmaxf(r.y, gb.y);
    float x2 = fminf(r.z, gb.z), y2 = fminf(r.w, gb.w);
    float inter = fmaxf(x2 - x1 + 1.f, 0.f) * fmaxf(y2 - y1 + 1.f, 0.f);
    float iou = __fdividef(inter, fmaxf(ar + ab - inter, 1e-6f));
    if (iou > best) { best = iou; bestn = base + i; }   // strict > : smallest n
  }
  if (vok) {
    u64 key = ((u64)__float_as_uint(fmaxf(best, 0.f)) << 32) | (u32)(~(u32)bestn);
    atomicMax(&bestkey[b * GSLOTS + lane], key);        // order-independent
  }
}

// ------------------------------------------------------------------
// Kernel C: fused assignment + losses with compacted GT lists,
// deterministic block reduction into per-block partials.
// ------------------------------------------------------------------
__global__ __launch_bounds__(BLK) void kloss(
    const float* __restrict__ cls,
    const float* __restrict__ bx, const float* __restrict__ by_,
    const float* __restrict__ bw, const float* __restrict__ bh,
    const float* __restrict__ rois,
    const float* __restrict__ gts_boxes,
    const int*   __restrict__ gts_cls,
    const int*   __restrict__ gts_val,
    const int*   __restrict__ gts_ign,
    const u64*   __restrict__ bestkey,
    float*       __restrict__ partials)
{
  __shared__ float4 sgt[GSLOTS];          // full GT boxes (TDM)
  __shared__ int    scls[GSLOTS];         // full GT classes
  __shared__ float4 cvbox[GSLOTS];        // compacted valid boxes (asc. g)
  __shared__ float  cvab[GSLOTS];         // their areas
  __shared__ int    cvcls[GSLOTS];        // their classes
  __shared__ float4 cibox[GSLOTS];        // compacted ignore boxes
  __shared__ int    tbn[GSLOTS], tg[GSLOTS]; // compacted take: anchor idx, g
  __shared__ int    scnt[3];              // nval, nign, ntake
  __shared__ float  sred[5][8];

  const int tid = threadIdx.x;
  const int b   = blockIdx.y;
  const int n   = blockIdx.x * BLK + tid;

#if HAVE_TDM
  if (tid < 32) tdm_load_512B(gts_boxes + (size_t)b * GSLOTS * 4, sgt);
#else
  if (tid < GSLOTS * 4)
    ((float*)sgt)[tid] = gts_boxes[(size_t)b * GSLOTS * 4 + tid];
#endif
  if (tid < GSLOTS) {                     // wave0: build compacted lists
    const int g  = tid;
    const int v  = gts_val[b * GSLOTS + g];
    const int ig = gts_ign[b * GSLOTS + g];
    const int cl = gts_cls[b * GSLOTS + g];
    scls[g] = cl;
    const u64 k  = bestkey[b * GSLOTS + g];
    const int bn = (int)(~(u32)k);
    const float bi = __uint_as_float((u32)(k >> 32));
    const bool pv = (v > 0);
    const bool pi = (ig > 0);
    const bool pt = pv && (k != 0ull) && (bi >= BEST_LO);
    const u32 vm = (u32)__ballot(pv);
    const u32 im = (u32)__ballot(pi);
    const u32 tm = (u32)__ballot(pt);
    const u32 lt = (1u << g) - 1u;
    const float4 gbv = sgt[g];            // TDM wave already waited tensorcnt
    if (pv) {
      int p = __popc(vm & lt);
      cvbox[p] = gbv;
      cvab[p]  = (gbv.z - gbv.x + 1.f) * (gbv.w - gbv.y + 1.f);
      cvcls[p] = cl;
    }
    if (pi) { int p = __popc(im & lt); cibox[p] = gbv; }
    if (pt) { int p = __popc(tm & lt); tbn[p] = bn; tg[p] = g; }
    if (g == 0) { scnt[0] = __popc(vm); scnt[1] = __popc(im); scnt[2] = __popc(tm); }
  }
  __syncthreads();

  // streaming prefetch of upcoming cls rows (global_prefetch_b8)
  if (n + 4 * BLK < N_ANCH)
    __builtin_prefetch(&cls[((size_t)b * N_ANCH + n + 4 * BLK) * NCLS], 0, 1);

  const int nv = scnt[0], ni = scnt[1], nt = scnt[2];

  const float4 r  = ((const float4*)rois)[n];
  const float rw  = r.z - r.x + 1.f, rh = r.w - r.y + 1.f;
  const float ar  = rw * rh;
  const float inv_ar = __frcp_rn(fmaxf(ar, 1e-6f));
  const float rcx = r.x + 0.5f * rw, rcy = r.y + 0.5f * rh;

  // argmax IoU over valid GTs (ascending g, strict > => first-max wins)
  float max_ol = -1.f;
  int   tj = 0;
  for (int j = 0; j < nv; ++j) {
    float4 gb = cvbox[j];
    float x1 = fmaxf(r.x, gb.x), y1 = fmaxf(r.y, gb.y);
    float x2 = fminf(r.z, gb.z), y2 = fminf(r.w, gb.w);
    float inter = fmaxf(x2 - x1 + 1.f, 0.f) * fmaxf(y2 - y1 + 1.f, 0.f);
    float iou = __fdividef(inter, fmaxf(ar + cvab[j] - inter, 1e-6f));
    if (iou > max_ol) { max_ol = iou; tj = j; }
  }
  // max inter/area(roi) over ignore GTs
  float max_ign = -1.f;
  for (int j = 0; j < ni; ++j) {
    float4 gb = cibox[j];
    float x1 = fmaxf(r.x, gb.x), y1 = fmaxf(r.y, gb.y);
    float x2 = fminf(r.z, gb.z), y2 = fminf(r.w, gb.w);
    float inter = fmaxf(x2 - x1 + 1.f, 0.f) * fmaxf(y2 - y1 + 1.f, 0.f);
    max_ign = fmaxf(max_ign, inter * inv_ar);
  }
  // best-anchor-per-GT scatter (ascending g => last-wins like XLA scatter)
  bool fgb = false; int tb = 0;
  for (int j = 0; j < nt; ++j)
    if (tbn[j] == n) { fgb = true; tb = tg[j]; }

  const bool fg     = (max_ol >= FG_LO) || fgb;
  const bool ign    = (!fg) && (max_ign >= IGN_LO);
  const bool active = fg || !ign;

  float4 gsel; int label;
  if (fgb)      { gsel = sgt[tb];   label = scls[tb]; }
  else if (fg)  { gsel = cvbox[tj]; label = cvcls[tj]; }
  else          { gsel = sgt[0];    label = 0; }

  // classification CE (log-softmax over C=4)
  const float4 cl = ((const float4*)cls)[(size_t)b * N_ANCH + n];
  float mx  = fmaxf(fmaxf(cl.x, cl.y), fmaxf(cl.z, cl.w));
  float lse = mx + logf(expf(cl.x - mx) + expf(cl.y - mx) +
                        expf(cl.z - mx) + expf(cl.w - mx));
  float clv = (label == 0) ? cl.x : (label == 1) ? cl.y : (label == 2) ? cl.z : cl.w;
  float ce  = lse - clv;

  float sl1 = 0.f, iouL = 0.f;
  if (fg) {
    float gw = gsel.z - gsel.x + 1.f, gh = gsel.w - gsel.y + 1.f;
    float gcx = gsel.x + 0.5f * gw, gcy = gsel.y + 0.5f * gh;
    // normalized targets (means=0, stds=[.1,.1,.2,.2])
    float tx = ((gcx - rcx) / rw) * 10.f;
    float ty = ((gcy - rcy) / rh) * 10.f;
    float tw = logf(fmaxf(gw, 1.f) / rw) * 5.f;
    float th = logf(fmaxf(gh, 1.f) / rh) * 5.f;
    size_t off = (size_t)b * N_ANCH + n;
    float dx = bx[off], dy = by_[off], dw = bw[off], dh = bh[off];
    float df;
    df = fabsf(dx - tx); sl1 += (df < 1.f) ? 0.5f * df * df : df - 0.5f;
    df = fabsf(dy - ty); sl1 += (df < 1.f) ? 0.5f * df * df : df - 0.5f;
    df = fabsf(dw - tw); sl1 += (df < 1.f) ? 0.5f * df * df : df - 0.5f;
    df = fabsf(dh - th); sl1 += (df < 1.f) ? 0.5f * df * df : df - 0.5f;
    // decode predicted box
    float pcx = rcx + (dx * 0.1f) * rw, pcy = rcy + (dy * 0.1f) * rh;
    float pw  = expf(dw * 0.2f) * rw,   ph  = expf(dh * 0.2f) * rh;
    float px1 = pcx - 0.5f * pw, py1 = pcy - 0.5f * ph;
    float px2 = pcx + 0.5f * pw - 1.f, py2 = pcy + 0.5f * ph - 1.f;
    // decode target box (gt with w,h clamped >= 1)
    float qw  = fmaxf(gw, 1.f), qh = fmaxf(gh, 1.f);
    float qx1 = gcx - 0.5f * qw, qy1 = gcy - 0.5f * qh;
    float qx2 = gcx + 0.5f * qw - 1.f, qy2 = gcy + 0.5f * qh - 1.f;
    float x1 = fmaxf(px1, qx1), y1 = fmaxf(py1, qy1);
    float x2 = fminf(px2, qx2), y2 = fminf(py2, qy2);
    float inter = fmaxf(x2 - x1 + 1.f, 0.f) * fmaxf(y2 - y1 + 1.f, 0.f);
    float pa = (px2 - px1 + 1.f) * (py2 - py1 + 1.f);
    float qa = (qx2 - qx1 + 1.f) * (qy2 - qy1 + 1.f);
    float i2 = inter / fmaxf(pa + qa - inter, 1e-6f);
    float sf = fminf(fmaxf(i2, 1e-6f), 1.f);
    iouL = -logf(sf);
  }

  float vals[5];
  vals[0] = active ? 1.f : 0.f;
  vals[1] = active ? ce  : 0.f;
  vals[2] = fg ? 1.f : 0.f;
  vals[3] = fg ? sl1  : 0.f;
  vals[4] = fg ? iouL : 0.f;

  const int lane = tid & 31, wv = tid >> 5;
  for (int j = 0; j < 5; ++j) {
    float v = wave_sum32_swz(vals[j]);         // ds_swizzle xor tree
    if (lane == 0) sred[j][wv] = v;
  }
  __syncthreads();
  if (tid == 0) {
    int bf = blockIdx.y * NBX + blockIdx.x;
    for (int j = 0; j < 5; ++j) {
      float s = 0.f;
      for (int w = 0; w < 8; ++w) s += sred[j][w];  // fixed order: deterministic
      partials[(size_t)j * NBLK + bf] = s;
    }
  }
}

// ------------------------------------------------------------------
// Kernel D: single-wave deterministic final combine (WMMA-assisted reduce).
// ------------------------------------------------------------------
__global__ void kfinal(const float* __restrict__ partials, float* __restrict__ out) {
  const int lane = threadIdx.x;          // 32 threads, EXEC all ones
  float s[5];
  for (int j = 0; j < 5; ++j) {
    float acc = 0.f;
    for (int i = lane; i < NBLK; i += 32)             // fixed per-lane order
      acc += partials[(size_t)j * NBLK + i];
    s[j] = wave_sum_wmma(acc);                        // v_wmma_f32_16x16x4_f32
  }
  if (lane == 0) {
    float cls_loss  = s[1] / fmaxf(s[0], 1.f);
    float nfg       = fmaxf(s[2], 1.f);
    float bbox_loss = s[3] / nfg;   // BBOX_2D_LAMBDA = 1
    float iou_loss  = s[4] / nfg;   // IOU_2D_LAMBDA = 1
    out[0] = cls_loss + bbox_loss + iou_loss;
  }
}

// ------------------------------------------------------------------
extern "C" void kernel_launch(void* const* d_in, const int* in_sizes, int n_in,
                              void* d_out, int out_size, void* d_ws, size_t ws_size,
                              hipStream_t stream) {
  const float* cls    = (const float*)d_in[0];
  // d_in[1] = prob (unused by reference)
  const float* bx     = (const float*)d_in[2];
  const float* by_    = (const float*)d_in[3];
  const float* bw     = (const float*)d_in[4];
  const float* bh     = (const float*)d_in[5];
  const float* rois   = (const float*)d_in[6];
  const float* gboxes = (const float*)d_in[7];
  const int*   gcls   = (const int*)d_in[8];
  const int*   gval   = (const int*)d_in[9];
  const int*   gign   = (const int*)d_in[10];

  u64*   bestkey  = (u64*)d_ws;
  float* partials = (float*)((char*)d_ws + (size_t)BATCH * GSLOTS * sizeof(u64));

  kinit<<<1, BATCH * GSLOTS, 0, stream>>>(bestkey);
  dim3 gridB(KB_BLOCKS, BATCH);
  kbest<<<gridB, BLK, 0, stream>>>(rois, gboxes, gval, bestkey);
  dim3 gridC(NBX, BATCH);
  kloss<<<gridC, BLK, 0, stream>>>(cls, bx, by_, bw, bh, rois, gboxes,
                                   gcls, gval, gign, bestkey, partials);
  kfinal<<<1, 32, 0, stream>>>(partials, (float*)d_out);
}